// TSUNegativeSampler_12859132084857
// MI455X (gfx1250) — compile-verified
//
#include <hip/hip_runtime.h>
#include <hip/hip_bf16.h>
#include <stdint.h>

// TSUNegativeSampler (top_k lowest-48, softmax(-E) categorical sample of 16)
// MI455X-tuned: single HBM read of the 1.05 GB energy tensor (~45us floor at
// 23.3 TB/s) by staging each 128 KB row into CDNA5's 320 KB/WGP LDS with
// global_load_async_to_lds_b128, then doing radix-style selection from LDS.

#define KSEL      48
#define NNEG      16
#define BLOCK     256
#define NBINS     4096
#define BIN_SHIFT 20
#define BELOW_CAP 64
#define BIN_CAP   512

// ---- order-preserving float<->u32 key (ascending float == ascending key) ----
__device__ __forceinline__ uint32_t f2key(float f) {
    uint32_t x = __float_as_uint(f);
    uint32_t mask = (uint32_t)((int32_t)x >> 31) | 0x80000000u;
    return x ^ mask;
}
__device__ __forceinline__ float key2f(uint32_t k) {
    uint32_t mask = (k & 0x80000000u) ? 0x80000000u : 0xFFFFFFFFu;
    return __uint_as_float(k ^ mask);
}

// LDS flat pointer -> wave-relative LDS byte offset (low 32 bits of aperture addr)
__device__ __forceinline__ uint32_t lds_off(const void* p) {
    return (uint32_t)(uintptr_t)p;
}

// ---- CDNA5 async global->LDS DMA (ASYNCcnt-tracked) ----
__device__ __forceinline__ void async_ld_b128(uint32_t ldsoff, uint64_t gaddr) {
    asm volatile("global_load_async_to_lds_b128 %0, %1, off"
                 :: "v"(ldsoff), "v"(gaddr) : "memory");
}
__device__ __forceinline__ void async_ld_b32(uint32_t ldsoff, uint64_t gaddr) {
    asm volatile("global_load_async_to_lds_b32 %0, %1, off"
                 :: "v"(ldsoff), "v"(gaddr) : "memory");
}
__device__ __forceinline__ void wait_async0() {
    asm volatile("s_wait_asynccnt 0x0" ::: "memory");
}

// ---- deterministic counter-based RNG (splitmix64) ----
__device__ __forceinline__ uint64_t mix64(uint64_t x) {
    x += 0x9E3779B97F4A7C15ull;
    x = (x ^ (x >> 30)) * 0xBF58476D1CE4E5B9ull;
    x = (x ^ (x >> 27)) * 0x94D049BB133111EBull;
    return x ^ (x >> 31);
}

extern "C" __global__ __launch_bounds__(BLOCK)
void tsu_negsample_kernel(const float* __restrict__ energy,
                          const int*   __restrict__ target,
                          float*       __restrict__ out,
                          int V, int nRows)
{
    extern __shared__ unsigned char smem_raw[];      // dynamic: V floats (row cache)
    float* lrow = (float*)smem_raw;

    __shared__ uint32_t hist[NBINS];
    __shared__ uint32_t scanbuf[BLOCK];
    __shared__ uint32_t belowKey[BELOW_CAP], belowIdx[BELOW_CAP];
    __shared__ uint32_t binKey[BIN_CAP],     binIdx[BIN_CAP];
    __shared__ uint32_t finalIdx[64];
    __shared__ float    finalE[64];
    __shared__ float    cdf[64];
    __shared__ uint32_t s_pivot, s_below, ctrBelow, ctrBin;

    const int row = blockIdx.x;
    const int tid = threadIdx.x;
    if (row >= nRows) return;

    // ---------------- stage entire row into LDS (async DMA) ----------------
    const uint64_t gbase = (uint64_t)(uintptr_t)(energy) + (uint64_t)row * (uint64_t)V * 4ull;
    const uint32_t rowBytes = (uint32_t)V * 4u;
    const uint32_t nb16 = rowBytes & ~15u;
    const uint32_t lbase = lds_off(lrow);

    for (uint32_t o = (uint32_t)tid * 16u; o < nb16; o += BLOCK * 16u)
        async_ld_b128(lbase + o, gbase + o);
    const uint32_t rem = rowBytes - nb16;                   // 0..12, multiple of 4
    if ((uint32_t)tid * 4u < rem)
        async_ld_b32(lbase + nb16 + (uint32_t)tid * 4u, gbase + nb16 + (uint32_t)tid * 4u);

    // overlap DMA with histogram clear + counter init
    for (int b = tid; b < NBINS; b += BLOCK) hist[b] = 0u;
    if (tid == 0) { ctrBelow = 0u; ctrBin = 0u; }

    wait_async0();          // my waves' async copies landed in LDS
    __syncthreads();        // everyone's copies landed

    const int targ = target[row];

    // ---------------- pass A: 4096-bin histogram of sortable keys ----------------
    for (int base = tid * 4; base < V; base += BLOCK * 4) {
        if (base + 3 < V) {
            float4 v = *(const float4*)(lrow + base);
            float vals[4] = {v.x, v.y, v.z, v.w};
            #pragma unroll
            for (int j = 0; j < 4; ++j) {
                int idx = base + j;
                if (idx == targ) continue;                   // exclude target (== +inf)
                atomicAdd(&hist[f2key(vals[j]) >> BIN_SHIFT], 1u);
            }
        } else {
            for (int idx = base; idx < V; ++idx)
                if (idx != targ)
                    atomicAdd(&hist[f2key(lrow[idx]) >> BIN_SHIFT], 1u);
        }
    }
    __syncthreads();

    // ---------------- find pivot bin (rank-KSEL) via block scan ----------------
    uint32_t lsum = 0;
    #pragma unroll
    for (int j = 0; j < NBINS / BLOCK; ++j) lsum += hist[tid * (NBINS / BLOCK) + j];
    scanbuf[tid] = lsum;
    __syncthreads();
    for (int o = 1; o < BLOCK; o <<= 1) {                    // Hillis-Steele inclusive
        uint32_t v = scanbuf[tid];
        uint32_t a = (tid >= o) ? scanbuf[tid - o] : 0u;
        __syncthreads();
        scanbuf[tid] = v + a;
        __syncthreads();
    }
    const uint32_t incl = scanbuf[tid];
    const uint32_t excl = incl - lsum;
    if (excl < KSEL && incl >= KSEL) {                       // exactly one thread
        uint32_t cum = excl;
        #pragma unroll
        for (int j = 0; j < NBINS / BLOCK; ++j) {
            uint32_t c = hist[tid * (NBINS / BLOCK) + j];
            if (cum + c >= KSEL) { s_pivot = (uint32_t)(tid * (NBINS / BLOCK) + j); s_below = cum; break; }
            cum += c;
        }
    }
    __syncthreads();

    // ---------------- pass B: gather candidates from LDS ----------------
    const uint32_t pivot = s_pivot;
    for (int idx = tid; idx < V; idx += BLOCK) {
        if (idx == targ) continue;
        uint32_t k = f2key(lrow[idx]);
        uint32_t b = k >> BIN_SHIFT;
        if (b < pivot) {
            uint32_t p = atomicAdd(&ctrBelow, 1u);
            if (p < BELOW_CAP) { belowKey[p] = k; belowIdx[p] = (uint32_t)idx; }
        } else if (b == pivot) {
            uint32_t p = atomicAdd(&ctrBin, 1u);
            if (p < BIN_CAP)   { binKey[p] = k; binIdx[p] = (uint32_t)idx; }
        }
    }
    __syncthreads();

    // ---------------- sort pivot-bin list (odd-even transposition) ----------------
    const uint32_t m = (ctrBin < (uint32_t)BIN_CAP) ? ctrBin : (uint32_t)BIN_CAP;
    for (uint32_t it = 0; it < m; ++it) {
        uint32_t i = 2u * (uint32_t)tid + (it & 1u);
        if (i + 1u < m) {
            uint32_t a = binKey[i], b = binKey[i + 1];
            if (a > b) {
                binKey[i] = b; binKey[i + 1] = a;
                uint32_t t0 = binIdx[i]; binIdx[i] = binIdx[i + 1]; binIdx[i + 1] = t0;
            }
        }
        __syncthreads();
    }

    // ---------------- compose exact KSEL-smallest set ----------------
    const uint32_t nBelow  = (ctrBelow < (uint32_t)BELOW_CAP) ? ctrBelow : (uint32_t)BELOW_CAP;
    uint32_t remNeed = (KSEL > nBelow) ? ((uint32_t)KSEL - nBelow) : 0u;
    if (remNeed > m) remNeed = m;
    const uint32_t fc = (nBelow + remNeed < 64u) ? (nBelow + remNeed) : 64u;  // normally 48

    if ((uint32_t)tid < fc) {
        uint32_t kk, ii;
        if ((uint32_t)tid < nBelow) { kk = belowKey[tid];          ii = belowIdx[tid]; }
        else                        { kk = binKey[tid - nBelow];   ii = binIdx[tid - nBelow]; }
        finalIdx[tid] = ii;
        finalE[tid]   = key2f(kk);
    }
    __syncthreads();

    // ---------------- softmax(-E) CDF over the fc candidates ----------------
    if (tid == 0) {
        float mlog = -3.4e38f;
        for (uint32_t j = 0; j < fc; ++j) mlog = fmaxf(mlog, -finalE[j]);
        float acc = 0.f;
        for (uint32_t j = 0; j < fc; ++j) { acc += __expf(-finalE[j] - mlog); cdf[j] = acc; }
    }
    __syncthreads();

    // ---------------- draw NNEG samples (inverse-CDF, counter RNG) ----------------
    if (tid < NNEG && fc > 0u) {
        uint64_t h = mix64(((uint64_t)(uint32_t)row << 32) ^ (uint64_t)((uint32_t)tid * 0x9E37u + 42u));
        float u   = (float)(uint32_t)(h >> 40) * (1.0f / 16777216.0f);   // [0,1)
        float t   = u * cdf[fc - 1u];
        uint32_t j = 0;
        while (j + 1u < fc && cdf[j] <= t) ++j;
        size_t o = (size_t)row * NNEG + (size_t)tid;
        out[o] = (float)finalIdx[j];                               // negative_indices
        out[(size_t)nRows * NNEG + o] = finalE[j];                 // negative_energies
    }
}

extern "C" void kernel_launch(void* const* d_in, const int* in_sizes, int n_in,
                              void* d_out, int out_size, void* d_ws, size_t ws_size,
                              hipStream_t stream) {
    (void)n_in; (void)out_size; (void)d_ws; (void)ws_size;
    const float* energy = (const float*)d_in[0];
    const int*   target = (const int*)d_in[1];
    const int nRows = in_sizes[1];                 // B*S = 8192
    const int V     = in_sizes[0] / in_sizes[1];   // 32000
    const size_t dynLds = (((size_t)V * 4u) + 15u) & ~(size_t)15u;   // 128 KB row cache
    tsu_negsample_kernel<<<dim3((unsigned)nRows), dim3(BLOCK), dynLds, stream>>>(
        energy, target, (float*)d_out, V, nRows);
}